// MultiTaskLossWrapper_OS_88708254531780
// MI455X (gfx1250) — compile-verified
//
#include <hip/hip_runtime.h>
#include <hip/hip_bf16.h>
#include <math.h>

// ---------------------------------------------------------------------------
// MultiTaskLossWrapper for MI455X (gfx1250, wave32)
//   B = 4096 fixed by the reference harness.
//   Cox logsumexp row-sums done on the matrix unit: V_WMMA_F32_16X16X4_F32
//   (masked exp values as A, ones as B  ->  D accumulates row sums in f32).
// ---------------------------------------------------------------------------

#define B 4096

typedef __attribute__((ext_vector_type(2))) float v2f;
typedef __attribute__((ext_vector_type(8))) float v8f;

__constant__ float c_RISK[9] = {1.0f, 1.0f, 0.91f, 1.12f, 1.71f, 2.41f, 3.27f, 5.18f, 8.44f};

// meta bit layout: [3:0]=risk bucket, 4=age<40, 5=age>=40, 6=age<65, 7=age>=65,
//                  8=grade==0, 9=grade in {1,2}
#define MB_LT40 (1 << 4)
#define MB_GE40 (1 << 5)
#define MB_LT65 (1 << 6)
#define MB_GE65 (1 << 7)
#define MB_G0   (1 << 8)
#define MB_G12  (1 << 9)

__device__ __forceinline__ float softplus_f(float x) {
    // stable log(1+exp(x))
    return (x > 0.0f) ? (x + log1pf(expf(-x))) : log1pf(expf(x));
}

// --------------------------------------------------------------------------
// K0: global max of h3/h1, precompute e = exp(h - hmax), meta bits, and the
//     9x9 swish(alpha) tables for each beta. Single block, 1024 threads.
// --------------------------------------------------------------------------
__global__ __launch_bounds__(1024)
void k0_prep(const float* __restrict__ h3, const float* __restrict__ h1,
             const int* __restrict__ age, const int* __restrict__ grade,
             const float* __restrict__ beta1, const float* __restrict__ beta2,
             float* __restrict__ e3, float* __restrict__ e1,
             int* __restrict__ meta, float* __restrict__ alphaT,
             float* __restrict__ hmax) {
    __shared__ float red3[1024];
    __shared__ float red1[1024];
    const int tid = threadIdx.x;

    float m3 = -3.4e38f, m1 = -3.4e38f;
    for (int i = tid; i < B; i += 1024) {
        m3 = fmaxf(m3, h3[i]);
        m1 = fmaxf(m1, h1[i]);
    }
    red3[tid] = m3;
    red1[tid] = m1;
    __syncthreads();
    for (int s = 512; s > 0; s >>= 1) {
        if (tid < s) {
            red3[tid] = fmaxf(red3[tid], red3[tid + s]);
            red1[tid] = fmaxf(red1[tid], red1[tid + s]);
        }
        __syncthreads();
    }
    const float gm3 = red3[0];
    const float gm1 = red1[0];
    if (tid == 0) {
        hmax[0] = gm3;
        hmax[1] = gm1;
    }

    for (int i = tid; i < B; i += 1024) {
        e3[i] = expf(h3[i] - gm3);
        e1[i] = expf(h1[i] - gm1);
        const int a = age[i];
        const int g = grade[i];
        int m = min(a / 10, 8);
        m |= (a < 40) ? MB_LT40 : MB_GE40;
        m |= (a < 65) ? MB_LT65 : MB_GE65;
        if (g == 0) m |= MB_G0;
        if (g == 1 || g == 2) m |= MB_G12;
        meta[i] = m;
    }

    if (tid < 162) {
        const int t = tid % 81;
        const int ri = t / 9;
        const int rj = t % 9;
        const float x = (c_RISK[rj] - c_RISK[ri]) * 0.125f;
        const float beta = (tid < 81) ? beta1[0] : beta2[0];
        const float sg = 1.0f / (1.0f + expf(-beta * x));
        alphaT[tid] = x * sg; // swish(x, beta)
    }
}

// --------------------------------------------------------------------------
// K1: Cox per-row contributions via WMMA f32 16x16x4.
//   Each wave (wave32) owns a 16-row tile. Per step K=4: lane L builds its
//   two A elements (row M = L&15, K = (L>>4)*2 + {0,1}) as
//   A[m,k] = (st[j+k] >= st[i_base+m]) ? e[j+k] : 0, B = ones.
//   LDS reads are UNCONDITIONAL b64 loads (mask applied via cndmask on
//   registers) so EXEC stays all-1s with no predicated ds_loads in the loop.
//   After the j-loop, C/D layout => lanes 0 and 16 hold the 16 row sums in
//   accumulator components 0..7 (rows M=0..7 and M=8..15 respectively).
//   grid = 32 blocks x 256 threads (8 waves) = 256 row tiles of 16 rows.
// --------------------------------------------------------------------------
__global__ __launch_bounds__(256)
void k1_cox_wmma(const float* __restrict__ st, const float* __restrict__ e3,
                 const float* __restrict__ e1, const float* __restrict__ h3,
                 const float* __restrict__ h1, const float* __restrict__ censor,
                 const float* __restrict__ hmax,
                 float* __restrict__ row3, float* __restrict__ row1) {
    __shared__ __align__(16) float lst[B];
    __shared__ __align__(16) float le3[B];
    __shared__ __align__(16) float le1[B];
    const int tid = threadIdx.x;
    for (int i = tid; i < B; i += 256) {
        lst[i] = st[i];
        le3[i] = e3[i];
        le1[i] = e1[i];
    }
    __syncthreads();

    const int wave = tid >> 5;
    const int lane = tid & 31;
    const int tile = blockIdx.x * 8 + wave;
    const int ibase = tile * 16;
    const int m = lane & 15;
    const int kofs = (lane >> 4) << 1;   // 0 or 2; (j + kofs) is 8B aligned
    const float sti = lst[ibase + m];

    v8f acc3 = {};
    v8f acc1 = {};
    v2f bones;
    bones.x = 1.0f;
    bones.y = 1.0f;

#pragma unroll 4
    for (int j = 0; j < B; j += 4) {
        // unconditional vector LDS loads (ds_load_b64), then register selects
        const v2f sv  = *(const v2f*)&lst[j + kofs];
        const v2f e3v = *(const v2f*)&le3[j + kofs];
        const v2f e1v = *(const v2f*)&le1[j + kofs];
        const bool c0 = (sv.x >= sti);
        const bool c1 = (sv.y >= sti);
        v2f a3, a1;
        a3.x = c0 ? e3v.x : 0.0f;
        a3.y = c1 ? e3v.y : 0.0f;
        a1.x = c0 ? e1v.x : 0.0f;
        a1.y = c1 ? e1v.y : 0.0f;
        acc3 = __builtin_amdgcn_wmma_f32_16x16x4_f32(
            false, a3, false, bones, (short)0, acc3, false, false);
        acc1 = __builtin_amdgcn_wmma_f32_16x16x4_f32(
            false, a1, false, bones, (short)0, acc1, false, false);
    }

    const float gm3 = hmax[0];
    const float gm1 = hmax[1];
    if (lane == 0 || lane == 16) {
        const int rbase = ibase + ((lane == 16) ? 8 : 0);
#pragma unroll
        for (int c = 0; c < 8; ++c) {
            const int i = rbase + c;
            const float cen = censor[i];
            row3[i] = cen * (h3[i] - (gm3 + logf(acc3[c])));
            row1[i] = cen * (h1[i] - (gm1 + logf(acc1[c])));
        }
    }
}

// --------------------------------------------------------------------------
// K2: pair-rank losses (lgg + hgg) over i<j pairs, hazard3d only.
//   One thread per row i; inner loop over j from LDS (all LDS reads are
//   unconditional; only the softplus math is under the mask branch).
//   Deterministic fixed-order LDS tree reduction; one 4-float partial per
//   block. grid = 16 blocks x 256 threads.
// --------------------------------------------------------------------------
__global__ __launch_bounds__(256)
void k2_rank(const float* __restrict__ h3g, const int* __restrict__ metag,
             const float* __restrict__ alphaT, float* __restrict__ partials) {
    __shared__ __align__(16) float lh[B];
    __shared__ __align__(16) int lm[B];
    __shared__ float lt[162];
    __shared__ float r0[256], r1[256], r2[256], r3[256];

    const int tid = threadIdx.x;
    for (int i = tid; i < B; i += 256) {
        lh[i] = h3g[i];
        lm[i] = metag[i];
    }
    if (tid < 162) lt[tid] = alphaT[tid];
    __syncthreads();

    const int gi = blockIdx.x * 256 + tid;
    const float hi = lh[gi];
    const int mi = lm[gi];
    const int bi9 = (mi & 15) * 9;
    const bool i_lgg = (mi & MB_LT40) && (mi & MB_G0);
    const bool i_hgg = (mi & MB_LT65) && (mi & MB_G12);

    float sl = 0.0f, cl = 0.0f, sh = 0.0f, ch = 0.0f;
    if (i_lgg || i_hgg) {
        for (int j = gi + 1; j < B; ++j) {
            const int mj = lm[j];
            const float hj = lh[j];            // unconditional LDS read
            const bool lgg = i_lgg && (mj & MB_GE40) && (mj & MB_G0);
            const bool hgg = i_hgg && (mj & MB_GE65) && (mj & MB_G12);
            if (lgg || hgg) {
                const int bj = mj & 15;
                const float dh = hi - hj;
                if (lgg) {
                    sl += softplus_f(lt[bi9 + bj] * dh);
                    cl += 1.0f;
                }
                if (hgg) {
                    sh += softplus_f(lt[81 + bi9 + bj] * dh);
                    ch += 1.0f;
                }
            }
        }
    }

    r0[tid] = sl;
    r1[tid] = cl;
    r2[tid] = sh;
    r3[tid] = ch;
    __syncthreads();
    for (int s = 128; s > 0; s >>= 1) {
        if (tid < s) {
            r0[tid] += r0[tid + s];
            r1[tid] += r1[tid + s];
            r2[tid] += r2[tid + s];
            r3[tid] += r3[tid + s];
        }
        __syncthreads();
    }
    if (tid == 0) {
        partials[blockIdx.x * 4 + 0] = r0[0];
        partials[blockIdx.x * 4 + 1] = r1[0];
        partials[blockIdx.x * 4 + 2] = r2[0];
        partials[blockIdx.x * 4 + 3] = r3[0];
    }
}

// --------------------------------------------------------------------------
// K3: deterministic final combine. Single block, 256 threads.
// --------------------------------------------------------------------------
__global__ __launch_bounds__(256)
void k3_final(const float* __restrict__ row3, const float* __restrict__ row1,
              const float* __restrict__ partials, const float* __restrict__ vars,
              float* __restrict__ out) {
    __shared__ float a3[256], a1[256];
    const int tid = threadIdx.x;
    float s3 = 0.0f, s1 = 0.0f;
    for (int i = tid; i < B; i += 256) {
        s3 += row3[i];
        s1 += row1[i];
    }
    a3[tid] = s3;
    a1[tid] = s1;
    __syncthreads();
    for (int s = 128; s > 0; s >>= 1) {
        if (tid < s) {
            a3[tid] += a3[tid + s];
            a1[tid] += a1[tid + s];
        }
        __syncthreads();
    }
    if (tid == 0) {
        const float loss3d = -a3[0] / (float)B;
        const float loss1d = -a1[0] / (float)B;
        float sl = 0.0f, cl = 0.0f, shh = 0.0f, chh = 0.0f;
        for (int b = 0; b < 16; ++b) {
            sl += partials[b * 4 + 0];
            cl += partials[b * 4 + 1];
            shh += partials[b * 4 + 2];
            chh += partials[b * 4 + 3];
        }
        const float lgg = (cl > 0.0f) ? (sl / cl) : 0.0f;
        const float hgg = (chh > 0.0f) ? (shh / chh) : 0.0f;
        const float losscli = lgg + hgg;
        const float v0 = vars[0], v2 = vars[2], v3 = vars[3];
        const float r = 0.5f * loss3d / (v0 * v0) + logf(v0) +
                        0.5f * loss1d / (v2 * v2) + logf(v2) +
                        0.5f * losscli / (v3 * v3) + logf(v3);
        out[0] = r;
    }
}

// --------------------------------------------------------------------------
// Workspace layout (floats):
//   [0      .. 4095 ]  e3
//   [4096   .. 8191 ]  e1
//   [8192   .. 12287]  row3 (per-row cox contributions, hazard3d)
//   [12288  .. 16383]  row1
//   [16384  .. 16545]  alpha tables (81 beta1 + 81 beta2)
//   [16546  .. 16547]  hmax3, hmax1
//   [16548  .. 16611]  rank partials (16 blocks x 4)
//   [16640  .. 20735]  meta (as int)
// --------------------------------------------------------------------------
extern "C" void kernel_launch(void* const* d_in, const int* in_sizes, int n_in,
                              void* d_out, int out_size, void* d_ws, size_t ws_size,
                              hipStream_t stream) {
    (void)in_sizes; (void)n_in; (void)out_size; (void)ws_size;
    const float* h3     = (const float*)d_in[0];
    const float* h1     = (const float*)d_in[1];
    const float* st     = (const float*)d_in[2];
    const float* censor = (const float*)d_in[3];
    const float* vars   = (const float*)d_in[4];
    const float* beta1  = (const float*)d_in[5];
    const float* beta2  = (const float*)d_in[6];
    const int*   age    = (const int*)d_in[7];
    const int*   grade  = (const int*)d_in[8];

    float* wsf      = (float*)d_ws;
    float* e3       = wsf;
    float* e1       = wsf + 4096;
    float* row3     = wsf + 8192;
    float* row1     = wsf + 12288;
    float* alphaT   = wsf + 16384;
    float* hmax     = wsf + 16546;
    float* partials = wsf + 16548;
    int*   meta     = (int*)(wsf + 16640);

    float* out = (float*)d_out;

    k0_prep<<<1, 1024, 0, stream>>>(h3, h1, age, grade, beta1, beta2,
                                    e3, e1, meta, alphaT, hmax);
    k1_cox_wmma<<<32, 256, 0, stream>>>(st, e3, e1, h3, h1, censor, hmax,
                                        row3, row1);
    k2_rank<<<16, 256, 0, stream>>>(h3, meta, alphaT, partials);
    k3_final<<<1, 256, 0, stream>>>(row3, row1, partials, vars, out);
}